// Model_38173669327128
// MI455X (gfx1250) — compile-verified
//
#include <hip/hip_runtime.h>

// ---------------------------------------------------------------------------
// Fused per-relation attention pooling for gfx1250 (MI455X).
//   h:[4,B,512] -> reshape [4,B,8,64]; hidden=tanh(T@W1+b1); s=hidden.w2(+b2);
//   alpha=softmax_r(s); out[b,h,:]=sum_r alpha*T.   b2 cancels in softmax.
// One wave handles 16 flattened (b,h) rows (4KB contiguous per relation).
// fp32 WMMA 16x16x4 for the 64x64 MLP matmul, everything staged in LDS.
// ---------------------------------------------------------------------------

typedef __attribute__((ext_vector_type(2))) float v2f;
typedef __attribute__((ext_vector_type(4))) float v4f;
typedef __attribute__((ext_vector_type(8))) float v8f;

#define R_REL 4
#define XROWSTRIDE 68                    // floats; pad 64->68 for bank-conflict-free ds
#define XTILEF (16 * XROWSTRIDE)         // 1088 floats per 16x64 relation tile
#define XWAVEF (R_REL * XTILEF)          // 4352 floats per wave
#define W1F_COUNT 4096                   // 64x64 fragments, pre-swizzled
#define WAVES_PER_BLOCK 8
#define LDS_FLOATS (W1F_COUNT + WAVES_PER_BLOCK * XWAVEF + WAVES_PER_BLOCK * 64 * 2)
#define LDS_BYTES (LDS_FLOATS * 4)       // 159744 B (< 320 KB WGP LDS)

__device__ __forceinline__ float fast_tanh(float x) {
#if __has_builtin(__builtin_amdgcn_tanhf)
  return __builtin_amdgcn_tanhf(x);
#else
  float r;
  // gfx1250 transcendental; v_nop covers the TRANS-op result hazard.
  asm("v_tanh_f32 %0, %1\n\tv_nop" : "=v"(r) : "v"(x));
  return r;
#endif
}

__device__ __forceinline__ float fast_exp2(float x) { return __builtin_amdgcn_exp2f(x); }
__device__ __forceinline__ float fast_rcp(float x)  { return __builtin_amdgcn_rcpf(x); }

// Wave-local LDS producer->consumer ordering (stores and loads are same-wave).
__device__ __forceinline__ void wave_lds_sync() {
  asm volatile("s_wait_dscnt 0" ::: "memory");
}

__global__ __launch_bounds__(256) void
fused_rel_attn(const float* __restrict__ h,   // [4, B, 512]
               const float* __restrict__ W1,  // [64, 64] row-major (d, e)
               const float* __restrict__ b1,  // [64]
               const float* __restrict__ w2,  // [64]
               float* __restrict__ out,       // [B, 512]
               int nB) {
  extern __shared__ float smem[];
  float* W1Ffrag = smem;                                   // 4096 floats
  float* Xall    = smem + W1F_COUNT;                       // 8 * 4352
  float* scoresS = Xall + WAVES_PER_BLOCK * XWAVEF;        // 8 * 64
  float* alphaS  = scoresS + WAVES_PER_BLOCK * 64;         // 8 * 64

  const int tid  = threadIdx.x;
  const int lane = tid & 31;
  const int wave = tid >> 5;
  const int m16  = lane & 15;
  const int hi   = (lane >> 4) & 1;       // 0: lanes 0-15, 1: lanes 16-31

  // ---- Stage W1 into B-fragment order: frag(n,k) lane l holds
  //      v0=W1[4k+(hi?2:0)][16n+m16], v1=W1[4k+1+(hi?2:0)][16n+m16]
  for (int p = tid; p < 2048; p += 256) {
    int n  = p >> 9;
    int k  = (p >> 5) & 15;
    int l  = p & 31;
    int d0 = 4 * k + ((l & 16) ? 2 : 0);
    int e  = 16 * n + (l & 15);
    W1Ffrag[2 * p]     = W1[d0 * 64 + e];
    W1Ffrag[2 * p + 1] = W1[(d0 + 1) * 64 + e];
  }

  // ---- Stage this wave's 4 relation tiles (each 4KB contiguous in h).
  const long tile      = (long)blockIdx.x * WAVES_PER_BLOCK + wave;
  const long relStride = (long)nB * 512;
  float* Xw = Xall + wave * XWAVEF;
  for (int r = 0; r < R_REL; ++r) {
    const float* src = h + (long)r * relStride + tile * 1024;
    float* dstT = Xw + r * XTILEF;
#pragma unroll
    for (int i = 0; i < 8; ++i) {
      int idx = lane + 32 * i;          // float4 index 0..255
      int row = idx >> 4;
      int c4  = idx & 15;
      v4f val = *(const v4f*)(src + idx * 4);
      *(v4f*)(dstT + row * XROWSTRIDE + c4 * 4) = val;   // 272B row stride, 16B aligned
    }
  }
  __syncthreads();

  // ---- Score MLP: hidden = tanh(X*W1 + b1); sp += hidden * w2 (per lane col).
  float sp[R_REL][8];
#pragma unroll
  for (int r = 0; r < R_REL; ++r)
#pragma unroll
    for (int v = 0; v < 8; ++v) sp[r][v] = 0.0f;

  const int afragOff = m16 * XROWSTRIDE + hi * 2;  // + 4k per K-tile

  for (int n = 0; n < 4; ++n) {                    // N-tiles of the 64 hidden cols
    float bv = b1[16 * n + m16];
    float wv = w2[16 * n + m16];
    v2f bf[16];
#pragma unroll
    for (int k = 0; k < 16; ++k)                   // B-fragments, reused across r
      bf[k] = *(const v2f*)(W1Ffrag + ((n * 16 + k) * 32 + lane) * 2);
#pragma unroll
    for (int r = 0; r < R_REL; ++r) {
      v8f d;
#pragma unroll
      for (int v = 0; v < 8; ++v) d[v] = bv;       // fold b1 into accumulator
#pragma unroll
      for (int k = 0; k < 16; ++k) {
        v2f a = *(const v2f*)(Xw + r * XTILEF + 4 * k + afragOff);
        d = __builtin_amdgcn_wmma_f32_16x16x4_f32(
                false, a, false, bf[k], (short)0, d, false, false);
      }
#pragma unroll
      for (int v = 0; v < 8; ++v) sp[r][v] += fast_tanh(d[v]) * wv;
    }
  }

  // ---- Reduce the w2-dot over each 16-lane half (C/D layout: reg v, half hi -> row v+8*hi).
  float* scW = scoresS + wave * 64;
#pragma unroll
  for (int r = 0; r < R_REL; ++r)
#pragma unroll
    for (int v = 0; v < 8; ++v) {
      float s = sp[r][v];
      s += __shfl_xor(s, 1, 32);
      s += __shfl_xor(s, 2, 32);
      s += __shfl_xor(s, 4, 32);
      s += __shfl_xor(s, 8, 32);
      sp[r][v] = s;
    }
  if (m16 == 0) {
    int mbase = hi * 8;
#pragma unroll
    for (int r = 0; r < R_REL; ++r)
#pragma unroll
      for (int v = 0; v < 8; ++v) scW[r * 16 + mbase + v] = sp[r][v];
  }
  wave_lds_sync();

  // ---- Softmax over relations, per row (b2 is a common shift: cancels).
  float* alW = alphaS + wave * 64;
  if (lane < 16) {
    float s0 = scW[lane], s1 = scW[16 + lane], s2 = scW[32 + lane], s3 = scW[48 + lane];
    float mx = fmaxf(fmaxf(s0, s1), fmaxf(s2, s3));
    const float L2E = 1.44269504f;
    float e0 = fast_exp2((s0 - mx) * L2E);
    float e1 = fast_exp2((s1 - mx) * L2E);
    float e2 = fast_exp2((s2 - mx) * L2E);
    float e3 = fast_exp2((s3 - mx) * L2E);
    float inv = fast_rcp(e0 + e1 + e2 + e3);
    alW[lane]      = e0 * inv;
    alW[16 + lane] = e1 * inv;
    alW[32 + lane] = e2 * inv;
    alW[48 + lane] = e3 * inv;
  }
  wave_lds_sync();

  // ---- Weighted sum over relations; coalesced b64 stores (rows contiguous in out).
  float* dst = out + tile * 1024;
  const int c = lane * 2;
#pragma unroll
  for (int m = 0; m < 16; ++m) {
    float a0 = alW[m], a1 = alW[16 + m], a2 = alW[32 + m], a3 = alW[48 + m];
    v2f x0 = *(const v2f*)(Xw + 0 * XTILEF + m * XROWSTRIDE + c);
    v2f x1 = *(const v2f*)(Xw + 1 * XTILEF + m * XROWSTRIDE + c);
    v2f x2 = *(const v2f*)(Xw + 2 * XTILEF + m * XROWSTRIDE + c);
    v2f x3 = *(const v2f*)(Xw + 3 * XTILEF + m * XROWSTRIDE + c);
    v2f acc = a0 * x0 + a1 * x1 + a2 * x2 + a3 * x3;
    *(v2f*)(dst + m * 64 + c) = acc;
  }
}

extern "C" void kernel_launch(void* const* d_in, const int* in_sizes, int n_in,
                              void* d_out, int out_size, void* d_ws, size_t ws_size,
                              hipStream_t stream) {
  const float* h  = (const float*)d_in[0];  // [4, B, 512]
  const float* W1 = (const float*)d_in[1];  // [64, 64]
  const float* b1 = (const float*)d_in[2];  // [64]
  const float* w2 = (const float*)d_in[3];  // [64]
  // d_in[4] = b2: uniform shift across relations -> cancels in softmax.
  float* out = (float*)d_out;               // [B, 512]

  const int nB     = in_sizes[0] / (R_REL * 512);      // 65536
  const int tiles  = (nB * 8) / 16;                    // 32768 wave-tiles
  const int blocks = tiles / WAVES_PER_BLOCK;          // 4096

  (void)d_ws; (void)ws_size; (void)n_in; (void)out_size;

  static_assert(LDS_BYTES < 320 * 1024, "exceeds WGP LDS");
  hipFuncSetAttribute((const void*)fused_rel_attn,
                      hipFuncAttributeMaxDynamicSharedMemorySize, LDS_BYTES);
  fused_rel_attn<<<blocks, 256, LDS_BYTES, stream>>>(h, W1, b1, w2, out, nB);
}